// ScaledDotProductAttention_69922067579340
// MI455X (gfx1250) — compile-verified
//
#include <hip/hip_runtime.h>
#include <hip/hip_bf16.h>

typedef float v2f __attribute__((ext_vector_type(2)));
typedef float v8f __attribute__((ext_vector_type(8)));

#define LD 65          // LDS row stride (64 + 1 pad -> conflict-free column walks)
#define EPS 1e-5f

__device__ __forceinline__ v8f wmma_f32(v2f a, v2f b, v8f c) {
  // V_WMMA_F32_16X16X4_F32 : D[16,16] = A[16,4] x B[4,16] + C
  return __builtin_amdgcn_wmma_f32_16x16x4_f32(
      /*neg_a=*/false, a, /*neg_b=*/false, b,
      /*c_mod=*/(short)0, c, /*reuse_a=*/false, /*reuse_b=*/false);
}

__device__ __forceinline__ float elu1(float x) {
  return x > 0.f ? x + 1.f : __expf(x);   // elu(x)+1 == exp(x) for x<=0
}

// ---------------------------------------------------------------------------
// Kernel 1: chunked causal linear attention for one (b,h) per workgroup.
//   att[b, n, h*64 + v]  (written directly in the transposed layout the
//   projection consumes)
// ---------------------------------------------------------------------------
__global__ __launch_bounds__(128) void linattn_kernel(
    const float* __restrict__ q, const float* __restrict__ k,
    const float* __restrict__ v, const float* __restrict__ mask,
    float* __restrict__ att) {
  __shared__ float sQ[64 * LD];
  __shared__ float sK[64 * LD];
  __shared__ float sV[64 * LD];
  __shared__ float sA[64 * LD];   // masked Q K^T
  __shared__ float sS[64 * LD];   // running state S[e][v]
  __shared__ float sZ[64];        // running state Z[e]
  __shared__ float sDen[64];      // per-row (q . cumsum k) + eps

  const int bh   = blockIdx.x;          // 0..15  (b*8 + h)
  const int b    = bh >> 3;
  const int h    = bh & 7;
  const int tid  = threadIdx.x;         // 0..127
  const int lane = tid & 31;
  const int wv   = tid >> 5;            // wave 0..3
  const int half = lane >> 4;           // lane half
  const int l16  = lane & 15;
  const int mrow = wv * 16;             // this wave's 16-row strip
  const size_t base = (size_t)bh * 1024 * 64;

  for (int i = tid; i < 64 * LD; i += 128) sS[i] = 0.f;
  if (tid < 64) sZ[tid] = 0.f;
  __syncthreads();

  for (int t = 0; t < 16; ++t) {
    // ---- stage chunk (feature map applied on the fly) ----
    for (int i = tid; i < 4096; i += 128) {
      int r = i >> 6, c = i & 63;
      size_t g = base + (size_t)(t * 64 + r) * 64 + c;
      sQ[r * LD + c] = elu1(q[g]);
      sK[r * LD + c] = elu1(k[g]) + mask[g];
      sV[r * LD + c] = v[g];
    }
    __syncthreads();

    // ---- A = causal(Q K^T), store masked result to sA ----
    for (int nb = 0; nb < 64; nb += 16) {
      v8f acc = {};
      for (int kk = 0; kk < 64; kk += 4) {
        const int ka = kk + 2 * half;
        v2f a, bb;
        a.x  = sQ[(mrow + l16) * LD + ka];
        a.y  = sQ[(mrow + l16) * LD + ka + 1];
        bb.x = sK[(nb + l16) * LD + ka];      // B[k][n] = K[n][k]
        bb.y = sK[(nb + l16) * LD + ka + 1];
        acc = wmma_f32(a, bb, acc);
      }
#pragma unroll
      for (int j = 0; j < 8; ++j) {
        int m = mrow + j + 8 * half;
        int n = nb + l16;
        sA[m * LD + n] = (n <= m) ? acc[j] : 0.f;   // inclusive causal mask
      }
    }
    __syncthreads();

    // ---- deno_arg[m] = rowsum(A) + q_m . Z_prev + eps  (rows 0..63) ----
    if (tid < 64) {
      float s = 0.f;
      for (int i = 0; i < 64; ++i)
        s += sA[tid * LD + i] + sQ[tid * LD + i] * sZ[i];
      sDen[tid] = s + EPS;
    }

    // ---- num = A@V + Q@S_prev  (4 v-tiles per wave) ----
    v8f numt[4];
    for (int nb4 = 0; nb4 < 4; ++nb4) {
      const int nb = nb4 * 16;
      v8f acc = {};
      for (int kk = 0; kk < 64; kk += 4) {      // intra-chunk: A @ V
        const int ka = kk + 2 * half;
        v2f a, bb;
        a.x  = sA[(mrow + l16) * LD + ka];
        a.y  = sA[(mrow + l16) * LD + ka + 1];
        bb.x = sV[ka * LD + nb + l16];
        bb.y = sV[(ka + 1) * LD + nb + l16];
        acc = wmma_f32(a, bb, acc);
      }
      for (int kk = 0; kk < 64; kk += 4) {      // inter-chunk: Q @ S_prev
        const int ka = kk + 2 * half;
        v2f a, bb;
        a.x  = sQ[(mrow + l16) * LD + ka];
        a.y  = sQ[(mrow + l16) * LD + ka + 1];
        bb.x = sS[ka * LD + nb + l16];
        bb.y = sS[(ka + 1) * LD + nb + l16];
        acc = wmma_f32(a, bb, acc);
      }
      numt[nb4] = acc;
    }
    __syncthreads();   // sDen ready; all reads of old sS complete

    // ---- attention = num * deno_arg  (faithful to reference's num/deno) ----
    for (int nb4 = 0; nb4 < 4; ++nb4) {
      const int nb = nb4 * 16;
#pragma unroll
      for (int j = 0; j < 8; ++j) {
        int m = mrow + j + 8 * half;
        int col = nb + l16;
        att[((size_t)b * 1024 + t * 64 + m) * 512 + h * 64 + col] =
            numt[nb4][j] * sDen[m];
      }
    }

    // ---- S += K^T @ V  (wave owns e-rows [mrow, mrow+16)) ----
    for (int nb = 0; nb < 64; nb += 16) {
      v8f acc;
#pragma unroll
      for (int j = 0; j < 8; ++j)
        acc[j] = sS[(mrow + j + 8 * half) * LD + nb + l16];
      for (int kk = 0; kk < 64; kk += 4) {
        const int ka = kk + 2 * half;
        v2f a, bb;
        a.x  = sK[ka * LD + mrow + l16];        // A[e][n] = K[n][e]
        a.y  = sK[(ka + 1) * LD + mrow + l16];
        bb.x = sV[ka * LD + nb + l16];
        bb.y = sV[(ka + 1) * LD + nb + l16];
        acc = wmma_f32(a, bb, acc);
      }
#pragma unroll
      for (int j = 0; j < 8; ++j)
        sS[(mrow + j + 8 * half) * LD + nb + l16] = acc[j];
    }

    // ---- Z += column sums of K ----
    if (tid < 64) {
      float z = sZ[tid];
      for (int i = 0; i < 64; ++i) z += sK[i * LD + tid];
      sZ[tid] = z;
    }
    __syncthreads();   // protect sQ/sK/sV/sA reuse next chunk
  }
}

// ---------------------------------------------------------------------------
// Kernel 2: out[2048,512] = att[2048,512] @ W^T + bias
// ---------------------------------------------------------------------------
__global__ __launch_bounds__(128) void proj_kernel(
    const float* __restrict__ att, const float* __restrict__ W,
    const float* __restrict__ bias, float* __restrict__ out) {
  __shared__ float sX[64 * LD];   // att rows tile  [r][k]
  __shared__ float sW[64 * LD];   // W tile         [c][k]  (B[k][c] = W[c][k])

  const int tid  = threadIdx.x;
  const int lane = tid & 31;
  const int wv   = tid >> 5;
  const int half = lane >> 4;
  const int l16  = lane & 15;
  const int rbase = blockIdx.x * 64;   // row tile in [0,2048)
  const int cbase = blockIdx.y * 64;   // col tile in [0,512)
  const int mrow  = wv * 16;

  v8f acc[4] = {};

  for (int kb = 0; kb < 512; kb += 64) {
    __syncthreads();
    for (int i = tid; i < 4096; i += 128) {
      int r = i >> 6, c = i & 63;
      sX[r * LD + c] = att[(size_t)(rbase + r) * 512 + kb + c];
      sW[r * LD + c] = W[(size_t)(cbase + r) * 512 + kb + c];
    }
    __syncthreads();

    for (int nb4 = 0; nb4 < 4; ++nb4) {
      const int nb = nb4 * 16;
      for (int kk = 0; kk < 64; kk += 4) {
        const int ka = kk + 2 * half;
        v2f a, bb;
        a.x  = sX[(mrow + l16) * LD + ka];
        a.y  = sX[(mrow + l16) * LD + ka + 1];
        bb.x = sW[(nb + l16) * LD + ka];
        bb.y = sW[(nb + l16) * LD + ka + 1];
        acc[nb4] = wmma_f32(a, bb, acc[nb4]);
      }
    }
  }

  for (int nb4 = 0; nb4 < 4; ++nb4) {
    const int nb = nb4 * 16;
    float bv = bias[cbase + nb + l16];
#pragma unroll
    for (int j = 0; j < 8; ++j) {
      int m = mrow + j + 8 * half;
      out[(size_t)(rbase + m) * 512 + cbase + nb + l16] = acc[nb4][j] + bv;
    }
  }
}

extern "C" void kernel_launch(void* const* d_in, const int* in_sizes, int n_in,
                              void* d_out, int out_size, void* d_ws, size_t ws_size,
                              hipStream_t stream) {
  const float* q    = (const float*)d_in[0];
  const float* k    = (const float*)d_in[1];
  const float* v    = (const float*)d_in[2];
  const float* mask = (const float*)d_in[3];
  const float* W    = (const float*)d_in[4];
  const float* bias = (const float*)d_in[5];
  float* att = (float*)d_ws;              // [2,1024,512] fp32 = 4 MB scratch
  float* out = (float*)d_out;             // [2,1024,512] fp32

  linattn_kernel<<<16, 128, 0, stream>>>(q, k, v, mask, att);
  dim3 grid(32, 8);                        // (2048/64) x (512/64)
  proj_kernel<<<grid, 128, 0, stream>>>(att, W, bias, out);
}